// NMSFilter_83700322664661
// MI455X (gfx1250) — compile-verified
//
#include <hip/hip_runtime.h>
#include <cstdint>
#include <cstddef>

// ---------------------------------------------------------------------------
// NMS filter for MI455X (gfx1250).
//   grid  = 160 workgroups (one per (batch,class) problem)
//   block = 256 threads = 8 wave32 waves; each thread OWNS 8 sorted candidates
// Pipeline per block:
//   1) async global->LDS DMA of the 2048-score row (gfx1250 ASYNCcnt path)
//   2) in-LDS bitonic sort of (score, idx), descending
//   3) gather owned boxes/areas into REGISTERS (and LDS for pivot broadcast)
//   4) greedy NMS: sequential pivot loop; suppression fully in registers,
//      keep bits published via exclusive-owner byte stores; barrier only
//      when the pivot is alive (uniform branch)
//   5) scatter score*keep back to original index order from registers
// ---------------------------------------------------------------------------

#define NMS_B 8
#define NMS_C 20
#define NMS_N 2048
#define NMS_T 256
#define NMS_OWN (NMS_N / NMS_T)   // 8 candidates per thread
#define NMS_THR 0.45f
#define PRE_THR 0.005f

#if defined(__HIP_DEVICE_COMPILE__)
#if __has_builtin(__builtin_amdgcn_global_load_async_to_lds_b128) && \
    __has_builtin(__builtin_amdgcn_s_wait_asynccnt)
#define USE_ASYNC_LDS 1
#else
#define USE_ASYNC_LDS 0
#endif
#else
#define USE_ASYNC_LDS 0
#endif

#if USE_ASYNC_LDS
// Builtin signature (probe-confirmed): (v4i32 addrspace(1)*, v4i32
// addrspace(3)*, imm offset, imm cpol).
typedef int v4i __attribute__((vector_size(16)));
typedef __attribute__((address_space(1))) v4i as1_v4i;
typedef __attribute__((address_space(3))) v4i as3_v4i;

__device__ __forceinline__ void async_copy_b128(const void* gsrc, void* ldst) {
  // 16B per lane, global -> LDS, tracked by ASYNCcnt.
  __builtin_amdgcn_global_load_async_to_lds_b128(
      (as1_v4i*)const_cast<void*>(gsrc), (as3_v4i*)ldst,
      /*offset=*/0, /*cpol=*/0);
}
#endif

__global__ __launch_bounds__(NMS_T) void nms_filter_kernel(
    const float* __restrict__ bbs,   // [B, N, 4]  (x1,y1,x2,y2)
    const float* __restrict__ conf,  // [B, C, N]
    float* __restrict__ out) {       // [B, C, N]
  __shared__ __align__(16) float    s_score[NMS_N];       // sorted scores
  __shared__ unsigned               s_idx[NMS_N];         // sorted -> original
  __shared__ __align__(16) float4   s_box[NMS_N];         // boxes (pivot bcast)
  __shared__ float                  s_area[NMS_N];        // areas (pivot bcast)
  __shared__ unsigned char          s_keepb[NMS_T];       // published keep bits

  const int tid = (int)threadIdx.x;
  const int b   = (int)blockIdx.x / NMS_C;
  const int c   = (int)blockIdx.x % NMS_C;

  const float*  gconf = conf + ((size_t)b * NMS_C + c) * NMS_N;
  const float4* gbox  = (const float4*)(bbs + (size_t)b * NMS_N * 4);
  float*        gout  = out + ((size_t)b * NMS_C + c) * NMS_N;

  // ---- 1) stage score row into LDS (async DMA path) ----------------------
#if USE_ASYNC_LDS
  for (int t = tid; t < NMS_N / 4; t += NMS_T)
    async_copy_b128(gconf + 4 * t, &s_score[4 * t]);
  __builtin_amdgcn_s_wait_asynccnt(0);
#else
  for (int t = tid; t < NMS_N; t += NMS_T) s_score[t] = gconf[t];
#endif
  for (int t = tid; t < NMS_N; t += NMS_T) s_idx[t] = (unsigned)t;
  __syncthreads();

  // ---- 2) bitonic sort (descending by score), payload = original index ---
  for (int k = 2; k <= NMS_N; k <<= 1) {
    for (int j = k >> 1; j > 0; j >>= 1) {
      for (int t = tid; t < NMS_N; t += NMS_T) {
        const int ixj = t ^ j;
        if (ixj > t) {
          const bool descBlock = ((t & k) == 0);
          const float a = s_score[t];
          const float d = s_score[ixj];
          const bool doSwap = descBlock ? (a < d) : (a > d);
          if (doSwap) {
            s_score[t] = d;  s_score[ixj] = a;
            const unsigned ia = s_idx[t];
            s_idx[t] = s_idx[ixj];  s_idx[ixj] = ia;
          }
        }
      }
      __syncthreads();
    }
  }

  // ---- 3) gather owned candidates into registers (+ LDS for broadcast) ---
  const int base = tid * NMS_OWN;   // contiguous ownership: j = base + q
  float4   rb[NMS_OWN];
  float    ra[NMS_OWN];
  unsigned rkeep = 0u;              // private keep bits, bit q <-> j=base+q
#pragma unroll
  for (int q = 0; q < NMS_OWN; ++q) {
    const int j = base + q;
    const float4 v = gbox[s_idx[j]];
    const float  a = fmaxf(v.z - v.x, 0.0f) * fmaxf(v.w - v.y, 0.0f);
    rb[q] = v;
    ra[q] = a;
    s_box[j]  = v;
    s_area[j] = a;
    if (s_score[j] > PRE_THR) rkeep |= (1u << q);
  }
  s_keepb[tid] = (unsigned char)rkeep;
  __syncthreads();

  // ---- 4) greedy NMS: sequential pivots, register-resident suppression ---
  for (int i = 0; i < NMS_N - 1; ++i) {
    // Uniform LDS broadcast read -> uniform branch across the block.
    const unsigned kb = (unsigned)s_keepb[i >> 3];
    if (kb & (1u << (i & 7))) {
      const float4 bi = s_box[i];   // uniform broadcast reads
      const float  ai = s_area[i];
      // Only threads with owned candidates beyond the pivot have work.
      if (base + NMS_OWN > i) {
        const unsigned old = rkeep;
#pragma unroll
        for (int q = 0; q < NMS_OWN; ++q) {
          const int j = base + q;
          const float4 bj = rb[q];
          float iw = fminf(bi.z, bj.z) - fmaxf(bi.x, bj.x);
          float ih = fminf(bi.w, bj.w) - fmaxf(bi.y, bj.y);
          iw = fmaxf(iw, 0.0f);
          ih = fmaxf(ih, 0.0f);
          const float inter = iw * ih;
          // iou > thr  <=>  inter > thr * max(ai + aj - inter, 1e-12)
          const float den = fmaxf(ai + ra[q] - inter, 1e-12f);
          if ((j > i) && (inter > NMS_THR * den)) rkeep &= ~(1u << q);
        }
        if (rkeep != old) s_keepb[tid] = (unsigned char)rkeep;  // owner store
      }
      __syncthreads();  // uniform: all threads reach this when pivot alive
    }
    // Dead pivot: no writes occurred this iteration, safe to skip barrier.
  }

  // ---- 5) scatter score*keep back to original order (from registers) -----
#pragma unroll
  for (int q = 0; q < NMS_OWN; ++q) {
    const int j = base + q;
    gout[s_idx[j]] = ((rkeep >> q) & 1u) ? s_score[j] : 0.0f;
  }
}

extern "C" void kernel_launch(void* const* d_in, const int* in_sizes, int n_in,
                              void* d_out, int out_size, void* d_ws,
                              size_t ws_size, hipStream_t stream) {
  (void)in_sizes; (void)n_in; (void)out_size; (void)d_ws; (void)ws_size;
  const float* bbs  = (const float*)d_in[0];  // [8, 2048, 4]
  const float* conf = (const float*)d_in[1];  // [8, 20, 2048]
  float*       out  = (float*)d_out;          // [8, 20, 2048]
  nms_filter_kernel<<<dim3(NMS_B * NMS_C), dim3(NMS_T), 0, stream>>>(bbs, conf,
                                                                     out);
}